// Refine_14439680049169
// MI455X (gfx1250) — compile-verified
//
#include <hip/hip_runtime.h>

#ifndef __has_builtin
#define __has_builtin(x) 0
#endif

typedef unsigned int v4u __attribute__((ext_vector_type(4)));
typedef int          v8i __attribute__((ext_vector_type(8)));
typedef int          v4i __attribute__((ext_vector_type(4)));

#if __has_builtin(__builtin_amdgcn_tensor_load_to_lds)
#define USE_TDM 1
#else
#define USE_TDM 0
#endif

#define KSEL 5000
#define HQ   256
#define WQ   256
#define HIN  512
#define WIN  512
#define NPIX 65536   // HQ*WQ per batch

// ---------------------------------------------------------------------------
// Fused antialiased 2x downsample: separable 4-tap [1,3,3,1]/8 (edges /7).
// One block = one (batch, channel, 8-output-row tile). Input tile of
// 18 rows x 512 cols staged into LDS by the Tensor Data Mover (D# descriptor,
// TENSOR_LOAD_TO_LDS + s_wait_tensorcnt).
// channels 0..35 -> out; channel 36 -> err -> err_q in workspace.
// ---------------------------------------------------------------------------
__global__ __launch_bounds__(256) void resize_kernel(
    const float* __restrict__ alp, const float* __restrict__ fgr,
    const float* __restrict__ hid, const float* __restrict__ err,
    float* __restrict__ out, float* __restrict__ errq)
{
  __shared__ float tile[18][512];            // 36 KB
  const int tid = threadIdx.x;
  const int ty  = blockIdx.x;                // 0..31  (8 output rows each)
  const int c   = blockIdx.y;                // 0..36
  const int b   = blockIdx.z;                // 0..3

  const int oy0   = ty * 8;
  const int base  = 2 * oy0 - 1;             // first input row tap
  const int rlo   = base < 0 ? 0 : base;
  const int rhi   = (base + 17 > 511) ? 511 : (base + 17);
  const int nrows = rhi - rlo + 1;           // 17 or 18
  const int lr0   = rlo - base;              // 0 or 1: LDS row of first loaded row

  // Branchless source-plane selection on integer addresses (s_cselect_b64).
  const size_t pbytes = (size_t)HIN * WIN * 4;
  const int cf  = (c >= 1) ? c - 1 : 0;
  const int chn = (c >= 4) ? c - 4 : 0;
  const unsigned long long pa = (unsigned long long)alp + (size_t)b * pbytes;
  const unsigned long long pf = (unsigned long long)fgr + ((size_t)b * 3  + cf)  * pbytes;
  const unsigned long long ph = (unsigned long long)hid + ((size_t)b * 32 + chn) * pbytes;
  const unsigned long long pe = (unsigned long long)err + (size_t)b * pbytes;
  const unsigned long long pp = (c == 0) ? pa : (c < 4) ? pf : (c < 36) ? ph : pe;
  const float* plane = (const float*)pp;

#if USE_TDM
  if (tid == 0) {
    const unsigned long long ga = pp + (unsigned long long)rlo * WIN * 4;
    v4u g0;
    g0[0] = 1u;                                   // count=1, user descriptor
    g0[1] = (unsigned)(lr0 * 2048);               // lds_addr (tile at LDS offset 0)
    g0[2] = (unsigned)ga;                         // global_addr[31:0]
    g0[3] = (unsigned)((ga >> 32) & 0x1FFFFFFull) // global_addr[56:32]
          | (2u << 30);                           // type=2 (image)
    v8i g1;
    g1[0] = 0x00020000;                           // data_size=2 -> 4 bytes
    g1[1] = (int)(512u << 16);                    // tensor_dim0[15:0]=512
    g1[2] = (int)(512u << 16);                    // dim0 hi=0 | tensor_dim1[15:0]=512
    g1[3] = (int)(512u << 16);                    // dim1 hi=0 | tile_dim0=512
    g1[4] = nrows;                                // tile_dim1 | tile_dim2=0
    g1[5] = 512;                                  // tensor_dim0_stride lo32
    g1[6] = 0;                                    // stride hi | dim1_stride lo
    g1[7] = 0;
    v4i gz = (v4i)0;
#if __clang_major__ >= 23
    v8i g4 = (v8i)0;
    __builtin_amdgcn_tensor_load_to_lds(g0, g1, gz, gz, g4, 0);
#else
    __builtin_amdgcn_tensor_load_to_lds(g0, g1, gz, gz, 0);
#endif
#if __has_builtin(__builtin_amdgcn_s_wait_tensorcnt)
    __builtin_amdgcn_s_wait_tensorcnt(0);
#else
    asm volatile("s_wait_tensorcnt 0x0" ::: "memory");
#endif
  }
#else
  // Fallback: cooperative synchronous staging (float4 = 128-bit loads).
  for (int i = tid; i < nrows * 128; i += 256) {
    const int r = i >> 7, q = i & 127;
    ((float4*)&tile[lr0 + r][0])[q] =
        ((const float4*)(plane + (size_t)(rlo + r) * WIN))[q];
  }
#endif

  // Zero-fill the (at most one) out-of-image row so dropped taps read 0.
  if (lr0 == 1)         { for (int x = tid; x < 512; x += 256) tile[0][x]  = 0.f; }
  if (lr0 + nrows < 18) { for (int x = tid; x < 512; x += 256) tile[17][x] = 0.f; }
  __syncthreads();

  // Horizontal 4-tap once per LDS row (clamped index + 0/1 edge weight),
  // then vertical 4-tap combine. Norm: 1/8 interior, 1/7 at borders.
  const int ox = tid;
  const int xb = 2 * ox;
  const int xm = (ox > 0)   ? xb - 1 : 0;
  const int xp = (ox < 255) ? xb + 2 : 511;
  const float wm = (ox > 0)   ? 1.f : 0.f;
  const float wp = (ox < 255) ? 1.f : 0.f;

  float h[18];
#pragma unroll
  for (int r = 0; r < 18; ++r) {
    const float* row = tile[r];
    h[r] = wm * row[xm] + 3.f * (row[xb] + row[xb + 1]) + wp * row[xp];
  }

  const float wxn = (ox == 0 || ox == 255) ? (1.f / 7.f) : (1.f / 8.f);
  float* dplane = (c < 36) ? out + ((size_t)b * 37 + c) * NPIX
                           : errq + (size_t)b * NPIX;
#pragma unroll
  for (int j = 0; j < 8; ++j) {
    const int oy = oy0 + j;
    const float acc = h[2*j] + 3.f * (h[2*j + 1] + h[2*j + 2]) + h[2*j + 3];
    const float wyn = (oy == 0 || oy == 255) ? (1.f / 7.f) : (1.f / 8.f);
    dplane[oy * WQ + ox] = acc * wxn * wyn;
  }
}

// ---------------------------------------------------------------------------
// Whole top-K select in ONE kernel: one block per batch, 1024 threads.
// Float bits of err_q (all >= 0) are order-preserving as uint32.
//   phase 0: zero both histograms; build hi-16-bit histogram (atomics)
//   phase A: suffix-scan hi histogram -> pivot hi-bin + count above
//   phase B: lo-16-bit histogram restricted to pivot bin (atomics)
//   phase C: suffix-scan lo histogram -> exact pivot key + remaining count
//   phase D: mask write; pivot-equal ties broken by LOWEST flat index
//            (matches jax.lax.top_k) via block-wide prefix scan.
// ---------------------------------------------------------------------------
__global__ __launch_bounds__(1024) void select_kernel(
    const float* __restrict__ errq, unsigned* __restrict__ histHi,
    unsigned* __restrict__ histLo, float* __restrict__ out)
{
  __shared__ unsigned sh[1024];
  __shared__ unsigned sres[4];    // [0]=topBin [1]=above1 [2]=pivot [3]=remaining
  const int b = blockIdx.x, tid = threadIdx.x;
  unsigned*    hh = histHi + (size_t)b * 65536;
  unsigned*    hl = histLo + (size_t)b * 65536;
  const float* e  = errq   + (size_t)b * NPIX;
  const int base = tid * 64;                       // contiguous 64-elem run

  // ---- phase 0: zero histograms, then build hi histogram ----
  for (int i = tid; i < 65536; i += 1024) { hh[i] = 0u; hl[i] = 0u; }
  __threadfence();
  __syncthreads();
  for (int i = 0; i < 64; ++i) {
    const unsigned k = __float_as_uint(e[base + i]);
    atomicAdd(&hh[k >> 16], 1u);
  }
  __threadfence();
  __syncthreads();

  // ---- phase A: hi scan. chunk = 64 bins per thread ----
  unsigned s = 0;
  for (int i = 0; i < 64; ++i) s += hh[tid * 64 + i];
  sh[tid] = s;
  __syncthreads();
  for (int off = 1; off < 1024; off <<= 1) {       // suffix (right) inclusive scan
    const unsigned v = (tid + off < 1024) ? sh[tid + off] : 0u;
    __syncthreads();
    sh[tid] += v;
    __syncthreads();
  }
  const unsigned sufIncl = sh[tid];                // elems in chunks >= mine
  const unsigned suf     = sufIncl - s;            // elems in chunks above mine
  if ((unsigned)KSEL > suf && (unsigned)KSEL <= sufIncl) {   // one thread
    unsigned cum = suf;
    for (int j = 63; j >= 0; --j) {
      const unsigned hv = hh[tid * 64 + j];
      if (cum + hv >= (unsigned)KSEL) { sres[0] = tid * 64 + j; sres[1] = cum; break; }
      cum += hv;
    }
  }
  __syncthreads();
  const unsigned topBin = sres[0];
  const unsigned need1  = (unsigned)KSEL - sres[1];
  __syncthreads();

  // ---- phase B: lo histogram of pivot hi-bin ----
  for (int i = 0; i < 64; ++i) {
    const unsigned k = __float_as_uint(e[base + i]);
    if ((k >> 16) == topBin) atomicAdd(&hl[k & 0xFFFFu], 1u);
  }
  __threadfence();
  __syncthreads();

  // ---- phase C: lo scan ----
  unsigned s2 = 0;
  for (int i = 0; i < 64; ++i) s2 += hl[tid * 64 + i];
  sh[tid] = s2;
  __syncthreads();
  for (int off = 1; off < 1024; off <<= 1) {
    const unsigned v = (tid + off < 1024) ? sh[tid + off] : 0u;
    __syncthreads();
    sh[tid] += v;
    __syncthreads();
  }
  const unsigned sufIncl2 = sh[tid];
  const unsigned suf2     = sufIncl2 - s2;
  if (need1 > suf2 && need1 <= sufIncl2) {         // one thread
    unsigned cum = suf2;
    for (int j = 63; j >= 0; --j) {
      const unsigned hv = hl[tid * 64 + j];
      if (cum + hv >= need1) {
        sres[2] = (topBin << 16) | (unsigned)(tid * 64 + j);
        sres[3] = need1 - cum;
        break;
      }
      cum += hv;
    }
  }
  __syncthreads();
  const unsigned pivot = sres[2];
  const unsigned rem   = sres[3];
  __syncthreads();

  // ---- phase D: deterministic mask write ----
  unsigned c = 0;
  for (int i = 0; i < 64; ++i)
    c += (__float_as_uint(e[base + i]) == pivot) ? 1u : 0u;
  sh[tid] = c;
  __syncthreads();
  for (int off = 1; off < 1024; off <<= 1) {       // forward inclusive scan
    const unsigned v = (tid >= off) ? sh[tid - off] : 0u;
    __syncthreads();
    sh[tid] += v;
    __syncthreads();
  }
  unsigned rank = sh[tid] - c;                     // exclusive prefix
  float* o = out + ((size_t)b * 37 + 36) * NPIX;
  for (int i = 0; i < 64; ++i) {
    const unsigned k = __float_as_uint(e[base + i]);
    float sel;
    if (k > pivot)       sel = 1.f;
    else if (k == pivot) { sel = (rank < rem) ? 1.f : 0.f; ++rank; }
    else                 sel = 0.f;
    o[base + i] = sel;
  }
}

// ---------------------------------------------------------------------------
extern "C" void kernel_launch(void* const* d_in, const int* in_sizes, int n_in,
                              void* d_out, int out_size, void* d_ws, size_t ws_size,
                              hipStream_t stream)
{
  (void)in_sizes; (void)n_in; (void)out_size; (void)ws_size;
  // inputs: 0=src, 1=bck (unused), 2=alp, 3=fgr, 4=err, 5=hid
  const float* alp = (const float*)d_in[2];
  const float* fgr = (const float*)d_in[3];
  const float* err = (const float*)d_in[4];
  const float* hid = (const float*)d_in[5];
  float* out = (float*)d_out;

  float*    errq   = (float*)d_ws;                 // 4 * 65536 f32
  unsigned* histHi = (unsigned*)d_ws + 4 * NPIX;   // 4 * 65536 u32
  unsigned* histLo = histHi + 4 * NPIX;            // 4 * 65536 u32

  resize_kernel<<<dim3(32, 37, 4), 256, 0, stream>>>(alp, fgr, hid, err, out, errq);
  select_kernel<<<4, 1024, 0, stream>>>(errq, histHi, histLo, out);
}